// xLSTM_56221121905286
// MI455X (gfx1250) — compile-verified
//
#include <hip/hip_runtime.h>
#include <math.h>

// ---------------------------------------------------------------------------
// xLSTM (3 mLSTM layers) forward, persistent-kernel design for MI455X/gfx1250.
//   grid = 8 workgroups (one per head), 256 threads (8 wave32) each.
//   WG h owns head h: its q/k/v/o/i/f slices, its recurrent (c,n,m) state,
//   its 128-col slice of x_t and 64-col slices of u / out.  3 grid barriers per layer.
//   All GEMMs via V_WMMA_F32_16X16X4_F32 (fp32, batch padded 4->16 cols).
// ---------------------------------------------------------------------------

#define NWG     8
#define NTHR    256
#define SLEN    512
#define LLAYERS 3
#define BB      4
#define II      512
#define HH      8
#define DDIM    64
#define HDDIM   512
#define PDIM    1024
#define EPSV    1e-5f

typedef float v2f __attribute__((ext_vector_type(2)));
typedef float v8f __attribute__((ext_vector_type(8)));

// LDS pitches (padded to dodge bank conflicts on strided row access)
#define XN_P 516
#define XT_P 1028
#define D_P  68

// LDS layout (float offsets)
#define XN_OFF  0        // 4*516  : x_n, later reused for u (down input)
#define XT_OFF  2064     // 4*1028 : full x_t
#define XC_OFF  6176     // 4*1028 : full x_c = silu(conv(x_t))
#define Q_OFF   10288    // 4*68 per-head slices
#define K_OFF   10560
#define V_OFF   10832
#define O_OFF   11104
#define RT_OFF  11376
#define HV_OFF  11648
#define RED_OFF 11920    // 256 generic reduction scratch
#define NQ_OFF  12176    // 256 reduction scratch
#define MS_OFF  12432    // 64  misc (LN sums, i/f gates)
#define SM_TOTAL 12496   // ~50 KB

struct KParams {
  const float* in[26];
  float* out;
  float* ws;
};

__device__ __forceinline__ float sig_(float x) { return 1.0f / (1.0f + expf(-x)); }

// D[feat 0..15][batchcol 0..15] += W[n0+m][k] * X[n][k]  (i.e. X @ W^T, batch in cols)
// A = weight tile (lane%16 = feature row, vgpr/half = k), B = activations from LDS.
__device__ __forceinline__ v8f wmma_tile(const float* __restrict__ W, int ldw,
                                         const float* X, int ldx,
                                         int n0, int Kdim, int lane) {
  const int idx  = lane & 15;
  const int half = lane >> 4;
  const float* wp = W + (size_t)(n0 + idx) * (size_t)ldw + 2 * half;
  const float* xp = X + (idx & 3) * ldx + 2 * half;
  __builtin_prefetch(wp, 0, 1);                       // global_prefetch_b8
  v8f acc = {0.f, 0.f, 0.f, 0.f, 0.f, 0.f, 0.f, 0.f};
#pragma unroll 8
  for (int kk = 0; kk < Kdim; kk += 4) {
    v2f a, b;
    a.x = wp[kk]; a.y = wp[kk + 1];
    b.x = xp[kk]; b.y = xp[kk + 1];
    acc = __builtin_amdgcn_wmma_f32_16x16x4_f32(false, a, false, b,
                                                (short)0, acc, false, false);
  }
  return acc;
}

// Monotonic-counter grid barrier (8 WGs, all execute identical barrier count).
__device__ __forceinline__ void gbar(unsigned* cnt, unsigned target) {
  __syncthreads();
  if (threadIdx.x == 0) {
    __threadfence();                 // release
    atomicAdd(cnt, 1u);
    while (atomicAdd(cnt, 0u) < target) __builtin_amdgcn_s_sleep(2);
  }
  __syncthreads();
  __threadfence();                   // acquire (invalidate near caches)
}

__global__ void xlstm_init_cnt(unsigned* c) { if (threadIdx.x == 0) c[0] = 0u; }

__global__ __launch_bounds__(NTHR) void xlstm_main(KParams P) {
  const int tid  = threadIdx.x;
  const int lane = tid & 31;
  const int wave = tid >> 5;
  const int h    = blockIdx.x;           // owned head / column-slice id

  __shared__ float sm[SM_TOTAL];

  float*    wsf  = P.ws;
  unsigned* cnt  = (unsigned*)P.ws;
  float* ws_x    = wsf + 64;                               // (B,I)  current layer input/output
  float* ws_xt   = ws_x + BB * II;                         // (B,P)  x_t
  float* ws_u    = ws_xt + BB * PDIM;                      // (B,HD) pre-down activation
  float* st_c    = ws_u + BB * HDDIM;                      // L*H*B*D*D
  float* st_n    = st_c + (size_t)LLAYERS * HH * BB * DDIM * DDIM;
  float* st_m    = st_n + (size_t)LLAYERS * HH * BB * DDIM;

  // ---- init this head's recurrent state (c=0, n=1, m=0) ----
  for (int l = 0; l < LLAYERS; ++l) {
    const int cb = (l * HH + h) * BB * DDIM * DDIM;
    for (int idx = tid; idx < BB * DDIM * DDIM; idx += NTHR) st_c[cb + idx] = 0.0f;
    const int nb = (l * HH + h) * BB * DDIM;
    for (int idx = tid; idx < BB * DDIM; idx += NTHR) st_n[nb + idx] = 1.0f;
    if (tid < BB) st_m[(l * HH + h) * BB + tid] = 0.0f;
  }
  __syncthreads();

  unsigned epoch = 0;

  for (int s = 0; s < SLEN; ++s) {
    for (int l = 0; l < LLAYERS; ++l) {
      const float* xsrc   = (l == 0) ? (P.in[0] + (size_t)s * BB * II) : ws_x;
      const float* ln_g   = P.in[1]  + l * II;
      const float* ln_b   = P.in[2]  + l * II;
      const float* upl_w  = P.in[3]  + (size_t)l * PDIM * II;
      const float* upl_b  = P.in[4]  + l * PDIM;
      const float* upr_w  = P.in[5]  + (size_t)l * HDDIM * II;
      const float* upr_b  = P.in[6]  + l * HDDIM;
      const float* conv_w = P.in[7]  + l * 4;
      const float  conv_b = P.in[8][l];
      const float* wq_w   = P.in[9]  + (size_t)l * HDDIM * PDIM;
      const float* wq_b   = P.in[10] + l * HDDIM;
      const float* wk_w   = P.in[11] + (size_t)l * HDDIM * PDIM;
      const float* wk_b   = P.in[12] + l * HDDIM;
      const float* wv_w   = P.in[13] + (size_t)l * HDDIM * PDIM;
      const float* wv_b   = P.in[14] + l * HDDIM;
      const float* wo_w   = P.in[15] + (size_t)l * HDDIM * PDIM;
      const float* wo_b   = P.in[16] + l * HDDIM;
      const float* wi_w   = P.in[17] + (size_t)(l * HH + h) * PDIM;
      const float  wi_b   = P.in[18][l * HH + h];
      const float* wf_w   = P.in[19] + (size_t)(l * HH + h) * PDIM;
      const float  wf_b   = P.in[20][l * HH + h];
      const float* skip_w = P.in[21] + (size_t)l * HDDIM * PDIM;
      const float* gn_g   = P.in[22] + l * HDDIM;
      const float* gn_b   = P.in[23] + l * HDDIM;
      const float* down_w = P.in[24] + (size_t)l * II * HDDIM;
      const float* down_b = P.in[25] + l * II;

      // ---------- LayerNorm (computed redundantly per WG) ----------
      for (int idx = tid; idx < BB * II; idx += NTHR) {
        int b = idx >> 9, i = idx & (II - 1);
        sm[XN_OFF + b * XN_P + i] = xsrc[b * II + i];
      }
      __syncthreads();
      if (wave < BB) {
        int b = wave;
        float s1 = 0.f, s2 = 0.f;
        for (int i = lane; i < II; i += 32) {
          float v = sm[XN_OFF + b * XN_P + i];
          s1 += v; s2 += v * v;
        }
        for (int off = 16; off > 0; off >>= 1) {
          s1 += __shfl_down(s1, off, 32);
          s2 += __shfl_down(s2, off, 32);
        }
        if (lane == 0) { sm[MS_OFF + b] = s1; sm[MS_OFF + 4 + b] = s2; }
      }
      __syncthreads();
      for (int idx = tid; idx < BB * II; idx += NTHR) {
        int b = idx >> 9, i = idx & (II - 1);
        float mu  = sm[MS_OFF + b] * (1.0f / II);
        float var = sm[MS_OFF + 4 + b] * (1.0f / II) - mu * mu;
        float xv  = sm[XN_OFF + b * XN_P + i];
        sm[XN_OFF + b * XN_P + i] =
            (xv - mu) * rsqrtf(var + EPSV) * ln_g[i] + ln_b[i];
      }
      __syncthreads();

      // ---------- up projections: x_t slice (8 tiles) + r_t slice (4 tiles) ----------
      for (int t = wave; t < 12; t += 8) {
        if (t < 8) {
          int n0 = h * 128 + t * 16;
          v8f acc = wmma_tile(upl_w, II, &sm[XN_OFF], XN_P, n0, II, lane);
          int n = lane & 15, hf = lane >> 4;
          if (n < BB) {
#pragma unroll
            for (int r = 0; r < 8; ++r) {
              int feat = n0 + r + 8 * hf;
              ws_xt[n * PDIM + feat] = acc[r] + upl_b[feat];
            }
          }
        } else {
          int n0 = h * 64 + (t - 8) * 16;
          v8f acc = wmma_tile(upr_w, II, &sm[XN_OFF], XN_P, n0, II, lane);
          int n = lane & 15, hf = lane >> 4;
          if (n < BB) {
#pragma unroll
            for (int r = 0; r < 8; ++r) {
              int feat = n0 + r + 8 * hf;
              sm[RT_OFF + n * D_P + (feat - h * 64)] = acc[r] + upr_b[feat];
            }
          }
        }
      }
      gbar(cnt, (++epoch) * NWG);   // barrier A: x_t complete everywhere

      // ---------- load full x_t, causal conv + silu -> x_c ----------
      for (int idx = tid; idx < BB * PDIM; idx += NTHR) {
        int b = idx >> 10, p = idx & (PDIM - 1);
        sm[XT_OFF + b * XT_P + p] = ws_xt[b * PDIM + p];
      }
      __syncthreads();
      {
        float cw0 = conv_w[0], cw1 = conv_w[1], cw2 = conv_w[2], cw3 = conv_w[3];
        for (int idx = tid; idx < BB * PDIM; idx += NTHR) {
          int b = idx >> 10, p = idx & (PDIM - 1);
          const float* xr = &sm[XT_OFF + b * XT_P];
          float y = conv_b;
          if (p >= 3) y += cw0 * xr[p - 3];
          if (p >= 2) y += cw1 * xr[p - 2];
          if (p >= 1) y += cw2 * xr[p - 1];
          y += cw3 * xr[p];
          sm[XC_OFF + b * XT_P + p] = y * sig_(y);
        }
      }
      __syncthreads();

      // ---------- q/k/v/o head slices (16 tiles, K=1024) ----------
      for (int t = wave; t < 16; t += 8) {
        int proj = t >> 2, tt = t & 3;
        int n0 = h * 64 + tt * 16;
        const float* W; const float* bias; const float* Xs; int dst;
        if      (proj == 0) { W = wq_w; bias = wq_b; Xs = &sm[XC_OFF]; dst = Q_OFF; }
        else if (proj == 1) { W = wk_w; bias = wk_b; Xs = &sm[XC_OFF]; dst = K_OFF; }
        else if (proj == 2) { W = wv_w; bias = wv_b; Xs = &sm[XT_OFF]; dst = V_OFF; }
        else                { W = wo_w; bias = wo_b; Xs = &sm[XT_OFF]; dst = O_OFF; }
        v8f acc = wmma_tile(W, PDIM, Xs, XT_P, n0, PDIM, lane);
        int n = lane & 15, hf = lane >> 4;
        if (n < BB) {
#pragma unroll
          for (int r = 0; r < 8; ++r) {
            int feat = n0 + r + 8 * hf;
            float v = acc[r] + bias[feat];
            if (proj == 1) v *= 0.125f;        // 1/sqrt(D)
            if (proj == 3) v = sig_(v);        // o gate sigmoid
            sm[dst + n * D_P + (feat - h * 64)] = v;
          }
        }
      }
      // ---------- i / f gates (wave w -> (b = w&3, sel = w>>2)) ----------
      {
        int b = wave & 3, sel = wave >> 2;
        const float* wr = sel ? wf_w : wi_w;
        float acc = 0.f;
        for (int p = lane; p < PDIM; p += 32)
          acc += sm[XC_OFF + b * XT_P + p] * wr[p];
        for (int off = 16; off > 0; off >>= 1) acc += __shfl_down(acc, off, 32);
        if (lane == 0)
          sm[MS_OFF + 8 + sel * 4 + b] = acc + (sel ? wf_b : wi_b);
      }
      __syncthreads();

      // ---------- mLSTM state update: thread = (b, d), WG-local state ----------
      {
        int b = tid >> 6, d = tid & 63;
        float* stm = st_m + (l * HH + h) * BB;
        float* stn = st_n + (l * HH + h) * BB * DDIM;
        float* stc = st_c + (size_t)((l * HH + h) * BB) * DDIM * DDIM;

        float mold = stm[b];
        float iv = sm[MS_OFF + 8 + b];
        float fv = sm[MS_OFF + 12 + b];
        float mnew = fmaxf(fv + mold, iv);
        float ig = expf(iv - mnew);
        float fg = expf(fv - mnew + mold);

        float kk = sm[K_OFF + b * D_P + d];
        float qq = sm[Q_OFF + b * D_P + d];
        float nold = stn[b * DDIM + d];
        float nnew = fg * nold + ig * kk;
        stn[b * DDIM + d] = nnew;
        sm[NQ_OFF + b * DDIM + d] = nnew * qq;
        __syncthreads();
        if (d == 0) stm[b] = mnew;            // safe: all reads of m done pre-sync
        for (int r = 32; r > 0; r >>= 1) {
          if (d < r) sm[NQ_OFF + b * DDIM + d] += sm[NQ_OFF + b * DDIM + d + r];
          __syncthreads();
        }
        float den = fmaxf(fabsf(sm[NQ_OFF + b * DDIM]), 1.0f);
        __syncthreads();                       // NQ free for reuse below

        float* crow = stc + (size_t)(b * DDIM + d) * DDIM;
        float vb = sm[V_OFF + b * D_P + d];
        float num = 0.f;
#pragma unroll 4
        for (int e = 0; e < DDIM; e += 4) {
          float4 c4 = *(float4*)(crow + e);
          float k0 = sm[K_OFF + b * D_P + e + 0];
          float k1 = sm[K_OFF + b * D_P + e + 1];
          float k2 = sm[K_OFF + b * D_P + e + 2];
          float k3 = sm[K_OFF + b * D_P + e + 3];
          c4.x = fg * c4.x + ig * vb * k0;
          c4.y = fg * c4.y + ig * vb * k1;
          c4.z = fg * c4.z + ig * vb * k2;
          c4.w = fg * c4.w + ig * vb * k3;
          num += c4.x * sm[Q_OFF + b * D_P + e + 0];
          num += c4.y * sm[Q_OFF + b * D_P + e + 1];
          num += c4.z * sm[Q_OFF + b * D_P + e + 2];
          num += c4.w * sm[Q_OFF + b * D_P + e + 3];
          *(float4*)(crow + e) = c4;
        }
        float hval = sm[O_OFF + b * D_P + d] * (num / den);

        // ---- per-head GroupNorm over D=64 ----
        sm[NQ_OFF + b * DDIM + d]  = hval;
        sm[RED_OFF + b * DDIM + d] = hval * hval;
        __syncthreads();
        for (int r = 32; r > 0; r >>= 1) {
          if (d < r) {
            sm[NQ_OFF + b * DDIM + d]  += sm[NQ_OFF + b * DDIM + d + r];
            sm[RED_OFF + b * DDIM + d] += sm[RED_OFF + b * DDIM + d + r];
          }
          __syncthreads();
        }
        float mu  = sm[NQ_OFF + b * DDIM] * (1.0f / DDIM);
        float var = sm[RED_OFF + b * DDIM] * (1.0f / DDIM) - mu * mu;
        float gn  = (hval - mu) * rsqrtf(var + EPSV) * gn_g[h * DDIM + d]
                  + gn_b[h * DDIM + d];
        sm[HV_OFF + b * D_P + d] = gn;
      }
      __syncthreads();

      // ---------- skip projection + gated u, write global slice ----------
      for (int t = wave; t < 4; t += 8) {
        int n0 = h * 64 + t * 16;
        v8f acc = wmma_tile(skip_w, PDIM, &sm[XC_OFF], XT_P, n0, PDIM, lane);
        int n = lane & 15, hf = lane >> 4;
        if (n < BB) {
#pragma unroll
          for (int r = 0; r < 8; ++r) {
            int feat = n0 + r + 8 * hf, dl = feat - h * 64;
            float rtv = sm[RT_OFF + n * D_P + dl];
            float uval = (acc[r] + sm[HV_OFF + n * D_P + dl]) * (rtv * sig_(rtv));
            ws_u[n * HDDIM + feat] = uval;
          }
        }
      }
      gbar(cnt, (++epoch) * NWG);   // barrier C: u complete everywhere

      // ---------- down projection + residual ----------
      for (int idx = tid; idx < BB * HDDIM; idx += NTHR) {
        int b = idx >> 9, i = idx & (HDDIM - 1);
        sm[XN_OFF + b * XN_P + i] = ws_u[b * HDDIM + i];   // reuse XN region
      }
      __syncthreads();
      for (int t = wave; t < 4; t += 8) {
        int n0 = h * 64 + t * 16;
        v8f acc = wmma_tile(down_w, HDDIM, &sm[XN_OFF], XN_P, n0, HDDIM, lane);
        int n = lane & 15, hf = lane >> 4;
        if (n < BB) {
#pragma unroll
          for (int r = 0; r < 8; ++r) {
            int feat = n0 + r + 8 * hf;
            float val = acc[r] + down_b[feat] + xsrc[n * II + feat];
            ws_x[n * II + feat] = val;
            if (l == LLAYERS - 1)
              P.out[((size_t)s * BB + n) * II + feat] = val;
          }
        }
      }
      gbar(cnt, (++epoch) * NWG);   // barrier D: new x complete everywhere
    }
  }
}

extern "C" void kernel_launch(void* const* d_in, const int* in_sizes, int n_in,
                              void* d_out, int out_size, void* d_ws, size_t ws_size,
                              hipStream_t stream) {
  (void)in_sizes; (void)n_in; (void)out_size; (void)ws_size;
  KParams P;
  for (int i = 0; i < 26; ++i) P.in[i] = (const float*)d_in[i];
  P.out = (float*)d_out;
  P.ws  = (float*)d_ws;
  xlstm_init_cnt<<<1, 32, 0, stream>>>((unsigned*)d_ws);
  xlstm_main<<<NWG, NTHR, 0, stream>>>(P);
}